// PRISM_1743756722487
// MI455X (gfx1250) — compile-verified
//
#include <hip/hip_runtime.h>
#include <hip/hip_bf16.h>
#include <stdint.h>

// Scatter-mean (segment_sum / counts, masked overwrite of center) for MI455X.
//
// Memory plan:
//   d_out  : reused as the f32 sums accumulator [C, D] (51.2 MB -> stays
//            resident in MI455X's 192 MB L2), overwritten in-place by finalize.
//   d_ws   : counts[C] (u32) + present[C] (u32)  (~400 KB).
//
// CDNA5-specific paths used:
//   * global_load_async_to_lds_b128 th:TH_LOAD_NT  (async copy engine; NT so
//     the 268 MB input stream does not evict the L2-resident accumulator)
//   * s_wait_asynccnt, ds_load_b128 readback
//   * global_atomic_add_f32 / _u32 (no-return) into L2-resident sums
//   * non-temporal load/store for the streamed center/out arrays

typedef __attribute__((ext_vector_type(4))) float v4f;
typedef __attribute__((address_space(3))) float lds_f;

#define ROWS_PER_BLOCK 8   // rows batched per block (8 async copies in flight)
#define MAX_LANES      64  // D/4 lanes per row; D == 256 -> 64 lanes (2 waves)

// ---------------------------------------------------------------- init ------
__global__ void prism_init_kernel(float* __restrict__ sums,
                                  unsigned* __restrict__ counts,
                                  unsigned* __restrict__ present,
                                  int C, long total4) {
    long gid = (long)blockIdx.x * blockDim.x + threadIdx.x;
    if (gid < total4) {
        v4f z = {0.0f, 0.0f, 0.0f, 0.0f};
        ((v4f*)sums)[gid] = z;
    }
    if (gid < C) {
        counts[gid]  = 0u;
        present[gid] = 0u;
    }
}

// ---------------------------------------------------------------- mark ------
__global__ void prism_mark_kernel(const int* __restrict__ target,
                                  unsigned* __restrict__ present, int T) {
    int gid = blockIdx.x * blockDim.x + threadIdx.x;
    if (gid < T) present[target[gid]] = 1u;   // idempotent, race-benign
}

// ------------------------------------------------------------- scatter ------
// Grid: ceil(N/8) blocks; block: D/4 lanes (64 for D=256; 2 waves).
// Batch of 8 rows per block:
//   1) issue 8 async global->LDS b128 copies per lane (ASYNCcnt = 8 in flight,
//      th:TH_LOAD_NT keeps the stream out of the accumulator's L2 footprint)
//   2) one s_wait_asynccnt 0
//   3) drain: ds_load_b128 each slot, 4 global_atomic_add_f32 per lane per row
// No barrier needed: each lane consumes only its own LDS slots and ASYNCcnt
// is a per-wave counter.
__global__ void prism_scatter_kernel(const float* __restrict__ inputs_row,
                                     const int*   __restrict__ target_row,
                                     float* __restrict__ sums,
                                     unsigned* __restrict__ counts,
                                     int N, int D) {
    __shared__ alignas(16) float smem[ROWS_PER_BLOCK * MAX_LANES * 4]; // 8 KB

    const int base = blockIdx.x * ROWS_PER_BLOCK;
    const int c    = threadIdx.x;                  // chunk index within a row

    // ---- stage: launch all async copies back-to-back -----------------------
#pragma unroll
    for (int r = 0; r < ROWS_PER_BLOCK; ++r) {
        const int row = base + r;
        if (row < N) {
            const float* gp = inputs_row + (size_t)row * D + (size_t)c * 4;
            lds_f* lp = (lds_f*)&smem[((size_t)r * MAX_LANES + c) * 4];
            asm volatile("global_load_async_to_lds_b128 %0, %1, off th:TH_LOAD_NT"
                         :: "v"(lp), "v"(gp)
                         : "memory");
        }
    }

    // ---- single wait for the whole batch -----------------------------------
#if __has_builtin(__builtin_amdgcn_s_wait_asynccnt)
    __builtin_amdgcn_s_wait_asynccnt(0);
#else
    asm volatile("s_wait_asynccnt 0" ::: "memory");
#endif

    // ---- drain: accumulate into L2-resident sums ---------------------------
#pragma unroll
    for (int r = 0; r < ROWS_PER_BLOCK; ++r) {
        const int row = base + r;
        if (row < N) {
            const int t = target_row[row];         // uniform per wave
            v4f v = *(const v4f*)&smem[((size_t)r * MAX_LANES + c) * 4];
            float* dst = sums + (size_t)t * D + (size_t)c * 4;
            atomicAdd(dst + 0, v.x);               // global_atomic_add_f32
            atomicAdd(dst + 1, v.y);
            atomicAdd(dst + 2, v.z);
            atomicAdd(dst + 3, v.w);
            if (c == 0) atomicAdd(&counts[t], 1u); // global_atomic_add_u32
        }
    }
}

// ------------------------------------------------------------ finalize ------
// Grid: C blocks. Block: D/4 lanes. out currently holds sums; overwrite
// in-place:  out = (present && cnt>0) ? sums/cnt : center.
__global__ void prism_finalize_kernel(const float* __restrict__ center,
                                      const unsigned* __restrict__ counts,
                                      const unsigned* __restrict__ present,
                                      float* __restrict__ out,
                                      int D) {
    const int cls = blockIdx.x;
    const int ch  = threadIdx.x;
    const size_t off = (size_t)cls * D + (size_t)ch * 4;

    const unsigned cnt = counts[cls];
    const bool upd = (present[cls] != 0u) && (cnt > 0u);

    v4f s   = *(const v4f*)(out + off);                               // L2 hit
    v4f cen = __builtin_nontemporal_load((const v4f*)(center + off)); // stream

    v4f r;
    if (upd) {
        const float cf = (float)cnt;
        r = s / cf;                        // matches reference's sums/counts
    } else {
        r = cen;
    }
    __builtin_nontemporal_store(r, (v4f*)(out + off));                // stream
}

// -------------------------------------------------------------- launch ------
extern "C" void kernel_launch(void* const* d_in, const int* in_sizes, int n_in,
                              void* d_out, int out_size, void* d_ws, size_t ws_size,
                              hipStream_t stream) {
    const float* inputs_row = (const float*)d_in[0];   // [N, D] f32
    const float* center     = (const float*)d_in[1];   // [C, D] f32
    const int*   target_row = (const int*)  d_in[2];   // [N]    i32
    const int*   target     = (const int*)  d_in[3];   // [T]    i32

    const int N = in_sizes[2];
    const int D = in_sizes[0] / N;        // 256
    const int C = in_sizes[1] / D;        // 50001
    const int T = in_sizes[3];            // 4096

    float*    sums    = (float*)d_out;            // accumulator, then result
    unsigned* counts  = (unsigned*)d_ws;          // [C]
    unsigned* present = counts + C;               // [C]

    // 1) zero sums / counts / present (deterministic per call)
    const long total4 = (long)C * D / 4;
    const int initBlocks = (int)((total4 + 255) / 256);
    prism_init_kernel<<<initBlocks, 256, 0, stream>>>(sums, counts, present, C, total4);

    // 2) mark classes present in `target`
    prism_mark_kernel<<<(T + 255) / 256, 256, 0, stream>>>(target, present, T);

    // 3) streaming scatter-add: 8-row batches through the async copy engine,
    //    NT-hinted stream, L2-resident atomics
    const int scatterBlocks = (N + ROWS_PER_BLOCK - 1) / ROWS_PER_BLOCK;
    prism_scatter_kernel<<<scatterBlocks, D / 4, 0, stream>>>(
        inputs_row, target_row, sums, counts, N, D);

    // 4) mean + masked overwrite of center, in-place into d_out
    prism_finalize_kernel<<<C, D / 4, 0, stream>>>(center, counts, present,
                                                   sums, D);
}